// AugmentationPipeline_58308476010521
// MI455X (gfx1250) — compile-verified
//
#include <hip/hip_runtime.h>
#include <stdint.h>

// ---- Problem constants (match reference) ----
#define AUG_B 32768
#define AUG_L 512
#define MASK_RATIO_F 0.2f
#define CROP_RATIO_F 0.8f
#define REORDER_RATIO_F 0.3f
#define MIN_LEN_C 3
#define MIN_W_C 2
#define MAX_W_C 5
#define NBIN 4096

struct AugKeys {
  uint32_t km0, km1;   // mask key
  uint32_t kc0, kc1;   // crop key
  uint32_t k10, k11;   // reorder wsize key
  uint32_t k20, k21;   // reorder start key
  uint32_t k30, k31;   // reorder apply key
  uint32_t k40, k41;   // reorder perm key
};

// ---- JAX threefry2x32 (Random123 schedule, as in jax._src.prng) ----
__host__ __device__ static inline uint32_t tf_rotl(uint32_t x, int n) {
  return (x << n) | (x >> (32 - n));
}

__host__ __device__ static inline void threefry2x32(uint32_t k0, uint32_t k1,
                                                    uint32_t c0, uint32_t c1,
                                                    uint32_t& o0, uint32_t& o1) {
  uint32_t ks[3] = {k0, k1, k0 ^ k1 ^ 0x1BD11BDAu};
  uint32_t x0 = c0 + ks[0];
  uint32_t x1 = c1 + ks[1];
  const int rotA[4] = {13, 15, 26, 6};
  const int rotB[4] = {17, 29, 16, 24};
#pragma unroll
  for (int i = 0; i < 5; ++i) {
    const int* r = (i & 1) ? rotB : rotA;
#pragma unroll
    for (int j = 0; j < 4; ++j) {
      x0 += x1;
      x1 = tf_rotl(x1, r[j]);
      x1 ^= x0;
    }
    x0 += ks[(i + 1) % 3];
    x1 += ks[(i + 2) % 3] + (uint32_t)(i + 1);
  }
  o0 = x0;
  o1 = x1;
}

// raw 32 random bits at flat counter index of a size-2*half draw
// (JAX splits the iota counter array in half: lane0 = [0,half), lane1 = rest)
__device__ static inline uint32_t tf_bits(uint32_t k0, uint32_t k1,
                                          uint32_t flat, uint32_t half) {
  uint32_t c0, c1, y0, y1;
  const bool lo = (flat < half);
  if (lo) { c0 = flat; c1 = flat + half; }
  else    { c0 = flat - half; c1 = flat; }
  threefry2x32(k0, k1, c0, c1, y0, y1);
  return lo ? y0 : y1;
}

__device__ static inline float tf_uniform01(uint32_t k0, uint32_t k1,
                                            uint32_t flat, uint32_t half) {
  const uint32_t bits = tf_bits(k0, k1, flat, half);
  return __uint_as_float((bits >> 9) | 0x3f800000u) - 1.0f;
}

__global__ __launch_bounds__(AUG_L) void AugmentationPipeline_58308476010521_kernel(
    const int* __restrict__ seq, const int* __restrict__ lens,
    float* __restrict__ out, AugKeys K) {
  __shared__ __align__(16) int      s_seq[AUG_L];
  __shared__ __align__(16) uint32_t s_key[AUG_L];
  __shared__ int s_vidx[AUG_L];
  __shared__ int s_reord[AUG_L];
  __shared__ __align__(16) int s_hist[NBIN];
  __shared__ int s_w0[16];
  __shared__ int s_w1[16];
  __shared__ int s_T;
  __shared__ int s_cstart;
  __shared__ int s_clen;
  __shared__ int s_capply;

  const uint32_t row = blockIdx.x;
  const int i = threadIdx.x;
  const int lane = i & 31;
  const int wav = i >> 5;
  const int len = lens[row];

  // ---- CDNA5 async global->LDS stage of this row (ASYNCcnt path) ----
  if (i < AUG_L / 4) {
    unsigned lds_addr = (unsigned)(size_t)(&s_seq[i * 4]);
    unsigned goff = (row * (unsigned)AUG_L + (unsigned)i * 4u) * 4u;
    asm volatile("global_load_async_to_lds_b128 %0, %1, %2"
                 :: "v"(lds_addr), "v"(goff), "s"(seq)
                 : "memory");
  }

  // ---- per-position mask score while the async load is in flight ----
  // score float u = (r>>9)/2^23 is a monotone injection of the integer r>>9,
  // so ranking the 23-bit integer key is exactly equivalent (ties included).
  const uint32_t mask_half = (uint32_t)AUG_B * (uint32_t)AUG_L / 2u;
  const uint32_t mbits =
      tf_bits(K.km0, K.km1, row * (uint32_t)AUG_L + (uint32_t)i, mask_half) >> 9;

  asm volatile("s_wait_asynccnt 0x0" ::: "memory");
  __syncthreads();

  const int sv = s_seq[i];
  const bool valid = (i < len) && (sv != 0);
  const uint32_t key = valid ? mbits : 0x00FFFFFFu;  // 'inf' sentinel > any valid key
  s_key[i] = key;
  s_reord[i] = sv;

  // ---- wave32 ballot scan of valid flags ----
  const uint32_t bm = (uint32_t)__ballot(valid ? 1 : 0);
  const int wtot = __popc(bm);
  const int lexcl = __popc(bm & ((1u << lane) - 1u));
  if (lane == 0) s_w0[wav] = wtot;
  __syncthreads();
  if (i < 16) {
    int acc = 0;
    for (int w = 0; w < i; ++w) acc += s_w0[w];
    s_w1[i] = acc;
  }
  __syncthreads();
  const int n_valid = s_w1[15] + s_w0[15];
  const int excl = s_w1[wav] + lexcl;
  // valid_idx = argsort(where(valid, pos, L+pos))
  if (valid) s_vidx[excl] = i;
  else       s_vidx[n_valid + (i - excl)] = i;

  // ---- histogram radix-select of the n_mask-th smallest key ----
  {
    const uint4 z = {0u, 0u, 0u, 0u};
    ((uint4*)s_hist)[i * 2 + 0] = z;
    ((uint4*)s_hist)[i * 2 + 1] = z;
  }
  if (i == 0) s_T = -1;
  __syncthreads();
  const uint32_t mybin = min(key >> 11, (uint32_t)(NBIN - 1));
  atomicAdd(&s_hist[mybin], 1);
  __syncthreads();

  int n_mask = (int)((float)n_valid * MASK_RATIO_F);
  n_mask = n_mask < 1 ? 1 : n_mask;
  n_mask = n_mask > n_valid ? n_valid : n_mask;

  int cnt[8];
  {
    const uint4 h0 = ((const uint4*)s_hist)[i * 2 + 0];
    const uint4 h1 = ((const uint4*)s_hist)[i * 2 + 1];
    cnt[0] = (int)h0.x; cnt[1] = (int)h0.y; cnt[2] = (int)h0.z; cnt[3] = (int)h0.w;
    cnt[4] = (int)h1.x; cnt[5] = (int)h1.y; cnt[6] = (int)h1.z; cnt[7] = (int)h1.w;
  }
  int tsum = 0;
#pragma unroll
  for (int k = 0; k < 8; ++k) tsum += cnt[k];
  // wave inclusive scan of per-thread bin totals via shfl_up, then wave offsets
  int sc = tsum;
#pragma unroll
  for (int d = 1; d < 32; d <<= 1) {
    const int nv = __shfl_up(sc, (unsigned)d, 32);
    if (lane >= d) sc += nv;
  }
  const int wexcl = sc - tsum;
  if (lane == 31) s_w0[wav] = sc;
  __syncthreads();
  if (i < 16) {
    int acc = 0;
    for (int w = 0; w < i; ++w) acc += s_w0[w];
    s_w1[i] = acc;
  }
  __syncthreads();
  {
    int acc = s_w1[wav] + wexcl;  // exclusive cum count below my first bin
#pragma unroll
    for (int k = 0; k < 8; ++k) {
      if (n_valid > 0 && acc < n_mask && n_mask <= acc + cnt[k]) s_T = i * 8 + k;
      acc += cnt[k];
    }
  }
  __syncthreads();
  const int T = s_T;

  // exact stable rank only for the (rare) waves holding a threshold-bin key
  const bool below = ((int)mybin < T);
  const bool at = ((int)mybin == T);
  int rank_full = 0;
  if (__ballot(at ? 1 : 0) != 0ull) {
    const uint4* k4 = (const uint4*)s_key;
    for (int jj = 0; jj < AUG_L / 4; ++jj) {
      const uint4 v = k4[jj];  // ds_load_b128 broadcast
      const int j0 = jj * 4;
      rank_full += (v.x < key || (v.x == key && (j0 + 0) < i)) ? 1 : 0;
      rank_full += (v.y < key || (v.y == key && (j0 + 1) < i)) ? 1 : 0;
      rank_full += (v.z < key || (v.z == key && (j0 + 2) < i)) ? 1 : 0;
      rank_full += (v.w < key || (v.w == key && (j0 + 3) < i)) ? 1 : 0;
    }
  }
  const bool apply_m = (len > 1) && (n_valid > 0);
  const bool do_mask = valid && apply_m && (below || (at && rank_full < n_mask));

  const size_t BL = (size_t)AUG_B * (size_t)AUG_L;
  __builtin_nontemporal_store((float)(do_mask ? 0 : sv),
                              &out[(size_t)row * AUG_L + (size_t)i]);
  if (i == 0) __builtin_nontemporal_store((float)len, &out[BL + row]);

  // ---- row-uniform crop/reorder scalars: WAVE 0 ONLY ----
  // (avoids 15 redundant waves re-issuing the scalarized SALU threefry chains)
  if (i < 32) {
    // crop scalars -> LDS broadcast
    const float uc = tf_uniform01(K.kc0, K.kc1, row, (uint32_t)AUG_B / 2u);
    int crop_len = (int)((float)len * CROP_RATIO_F);
    crop_len = crop_len < MIN_LEN_C ? MIN_LEN_C : crop_len;
    crop_len = crop_len > len ? len : crop_len;
    int max_start = len - crop_len + 1;
    if (max_start < 1) max_start = 1;
    const int cstart = (int)(uc * (float)max_start);
    if (lane == 0) {
      s_cstart = cstart;
      s_clen = crop_len;
      s_capply = (len > MIN_LEN_C) ? 1 : 0;
    }

    // reorder window + permutation (uniform across wave 0)
    const uint32_t halfB = (uint32_t)AUG_B / 2u;
    const float u1 = tf_uniform01(K.k10, K.k11, row, halfB);
    const float u2 = tf_uniform01(K.k20, K.k21, row, halfB);
    const float u3 = tf_uniform01(K.k30, K.k31, row, halfB);
    float u4[MAX_W_C];
    const uint32_t half5 = (uint32_t)AUG_B * (uint32_t)MAX_W_C / 2u;
#pragma unroll
    for (int k = 0; k < MAX_W_C; ++k)
      u4[k] = tf_uniform01(K.k40, K.k41, row * (uint32_t)MAX_W_C + (uint32_t)k, half5);

    const int max_w = n_valid < MAX_W_C ? n_valid : MAX_W_C;
    int span = max_w - MIN_W_C + 1;
    if (span < 1) span = 1;
    int ws = MIN_W_C + (int)(u1 * (float)span);
    ws = ws < MIN_W_C ? MIN_W_C : (ws > MAX_W_C ? MAX_W_C : ws);
    int mstart = n_valid - ws + 1;
    if (mstart < 1) mstart = 1;
    const int rstart = (int)(u2 * (float)mstart);
    const bool apply_r =
        (len > MIN_W_C) && (u3 <= REORDER_RATIO_F) && (n_valid >= MIN_W_C);

    int wc[MAX_W_C], wi[MAX_W_C];
#pragma unroll
    for (int k = 0; k < MAX_W_C; ++k) {
      int s = rstart + k;
      if (s > AUG_L - 1) s = AUG_L - 1;
      wc[k] = s_vidx[s];      // uniform address -> LDS broadcast
      wi[k] = s_seq[wc[k]];
    }
    float kk[MAX_W_C];
#pragma unroll
    for (int k = 0; k < MAX_W_C; ++k)
      kk[k] = (k < ws) ? u4[k] : __builtin_inff();
    int perm[MAX_W_C] = {0, 1, 2, 3, 4};
    for (int a = 1; a < MAX_W_C; ++a) {
      const int p = perm[a];
      const float kv = kk[p];
      int b = a - 1;
      while (b >= 0 && kk[perm[b]] > kv) { perm[b + 1] = perm[b]; --b; }
      perm[b + 1] = p;
    }
    if (lane == 0) {
      for (int k = 0; k < MAX_W_C; ++k) {
        const int sh = wi[perm[k]];
        const int v = ((k < ws) && apply_r) ? sh : wi[k];
        s_reord[wc[k]] = v;   // sequential: last-write-wins on clamped dup cols
      }
    }
  }
  __syncthreads();

  // =================== crop augmentation store ===================
  {
    const int cstart = s_cstart;          // LDS broadcast reads
    const int crop_len = s_clen;
    const bool apply_c = (s_capply != 0);
    int gidx = cstart + i;
    if (gidx > AUG_L - 1) gidx = AUG_L - 1;
    const int cval = (i < crop_len) ? s_seq[gidx] : 0;
    __builtin_nontemporal_store(
        (float)(apply_c ? cval : sv),
        &out[BL + (size_t)AUG_B + (size_t)row * AUG_L + (size_t)i]);
    if (i == 0)
      __builtin_nontemporal_store((float)(apply_c ? crop_len : len),
                                  &out[2 * BL + (size_t)AUG_B + row]);
  }

  // =================== reorder augmentation store ===================
  __builtin_nontemporal_store(
      (float)s_reord[i],
      &out[2 * BL + 2 * (size_t)AUG_B + (size_t)row * AUG_L + (size_t)i]);
  if (i == 0)
    __builtin_nontemporal_store((float)len, &out[3 * BL + 2 * (size_t)AUG_B + row]);
}

extern "C" void kernel_launch(void* const* d_in, const int* in_sizes, int n_in,
                              void* d_out, int out_size, void* d_ws, size_t ws_size,
                              hipStream_t stream) {
  (void)in_sizes; (void)n_in; (void)out_size; (void)d_ws; (void)ws_size;
  const int* seq = (const int*)d_in[0];
  const int* lens = (const int*)d_in[1];
  float* out = (float*)d_out;

  // ---- Host-side JAX key derivation (deterministic, no HIP API calls) ----
  // root = key(42) = (0, 42); km, kc, kr = split(root, 3):
  // counts iota(6) split in half -> pairs (0,3),(1,4),(2,5);
  // flat bits = [y0(0,3), y0(1,4), y0(2,5), y1(0,3), y1(1,4), y1(2,5)]
  // km = (flat0, flat1), kc = (flat2, flat3), kr = (flat4, flat5)
  uint32_t a0, a1, b0, b1, c0, c1;
  threefry2x32(0u, 42u, 0u, 3u, a0, a1);
  threefry2x32(0u, 42u, 1u, 4u, b0, b1);
  threefry2x32(0u, 42u, 2u, 5u, c0, c1);
  const uint32_t km0 = a0, km1 = b0;
  const uint32_t kc0 = c0, kc1 = a1;
  const uint32_t kr0 = b1, kr1 = c1;

  // k1..k4 = split(kr, 4): pairs (0,4),(1,5),(2,6),(3,7);
  // flat = [p0,q0,s0,t0, p1,q1,s1,t1]; k1=(p0,q0) k2=(s0,t0) k3=(p1,q1) k4=(s1,t1)
  uint32_t p0, p1, q0, q1, s0, s1, t0, t1;
  threefry2x32(kr0, kr1, 0u, 4u, p0, p1);
  threefry2x32(kr0, kr1, 1u, 5u, q0, q1);
  threefry2x32(kr0, kr1, 2u, 6u, s0, s1);
  threefry2x32(kr0, kr1, 3u, 7u, t0, t1);

  AugKeys K;
  K.km0 = km0; K.km1 = km1;
  K.kc0 = kc0; K.kc1 = kc1;
  K.k10 = p0;  K.k11 = q0;
  K.k20 = s0;  K.k21 = t0;
  K.k30 = p1;  K.k31 = q1;
  K.k40 = s1;  K.k41 = t1;

  dim3 grid(AUG_B), block(AUG_L);
  hipLaunchKernelGGL(AugmentationPipeline_58308476010521_kernel,
                     grid, block, 0, stream, seq, lens, out, K);
}